// QCNNClassifier_54331336294815
// MI455X (gfx1250) — compile-verified
//
#include <hip/hip_runtime.h>
#include <math.h>

typedef float v2f __attribute__((ext_vector_type(2)));
typedef float v8f __attribute__((ext_vector_type(8)));

#define NQk   4
#define NLk   4
#define NFk   4
#define DIMk  16
#define HIDk  64
#define BATCHk 16384
#define PI_F  3.14159265358979f

// workspace layout (in floats)
static constexpr int WS_BQ     = 0;                       // 4*2*8*32*2 = 4096
static constexpr int WS_BW1    = 4096;                    // 4*16*32*2  = 4096
static constexpr int WS_FEAT   = 8192;                    // BATCH*64
static constexpr int WS_H      = 8192 + BATCHk * 64;      // BATCH*64
static constexpr int WS_COEF   = 8192 + 2 * BATCHk * 64;  // 64
static constexpr int WS_DCONST = WS_COEF + 64;            // 1

// ---------------------------------------------------------------------------
// Kernel A: simulate the 4-qubit circuit per column to build U_f (16x16
// complex, 4 filters), then pre-swizzle the WMMA B-operands for the quantum
// matvec (Bq) and for the W1 GEMM (Bw1) into the per-lane f32 4x16 B layout:
//   B-pair VGPR0 : lanes 0-15 -> local K0, lanes 16-31 -> local K2
//   B-pair VGPR1 : lanes 0-15 -> local K1, lanes 16-31 -> local K3
// ---------------------------------------------------------------------------
__global__ void __launch_bounds__(64)
build_tables(const float* __restrict__ qw,   // (NF, NL, 8)
             const float* __restrict__ W1,   // (64, 64)
             float* __restrict__ ws) {
  __shared__ float Ure[NFk][DIMk][DIMk];
  __shared__ float Uim[NFk][DIMk][DIMk];
  const int t = threadIdx.x;      // 64 threads: (filter, column)
  const int f = t >> 4;
  const int col = t & 15;

  float re[16], im[16];
#pragma unroll
  for (int b = 0; b < 16; ++b) { re[b] = (b == col) ? 1.f : 0.f; im[b] = 0.f; }

  for (int l = 0; l < NLk; ++l) {
    const float* w = qw + (f * NLk + l) * 8;
    // RX on wires 0..3  (wire w <-> state bit (3-w), bit3 = MSB)
#pragma unroll
    for (int i = 0; i < 4; ++i) {
      const float h = 0.5f * w[i];
      const float c = cosf(h), s = sinf(h);
      const int mask = 1 << (3 - i);
#pragma unroll
      for (int b = 0; b < 16; ++b) {
        if (b & mask) continue;
        const int b1 = b | mask;
        const float r0 = re[b], i0 = im[b], r1 = re[b1], i1 = im[b1];
        re[b]  = c * r0 + s * i1;   // c*a0 + (-i s)*a1
        im[b]  = c * i0 - s * r1;
        re[b1] = c * r1 + s * i0;   // (-i s)*a0 + c*a1
        im[b1] = c * i1 - s * r0;
      }
    }
    // RZ on wires 0..3: bit=0 -> e^{-it/2}, bit=1 -> e^{+it/2}
#pragma unroll
    for (int i = 0; i < 4; ++i) {
      const float h = 0.5f * w[4 + i];
      const float c = cosf(h), s = sinf(h);
      const int mask = 1 << (3 - i);
#pragma unroll
      for (int b = 0; b < 16; ++b) {
        const float ci = (b & mask) ? s : -s;
        const float r = re[b], q = im[b];
        re[b] = r * c - q * ci;
        im[b] = r * ci + q * c;
      }
    }
    // CNOT(3,2),(2,1),(1,0): (ctrl bit, tgt bit) = (0,1),(1,2),(2,3)
#pragma unroll
    for (int k = 0; k < 3; ++k) {
      const int cm = 1 << k, tm = 1 << (k + 1);
#pragma unroll
      for (int b = 0; b < 16; ++b) {
        if ((b & cm) && !(b & tm)) {
          const int b1 = b | tm;
          float tr = re[b]; re[b] = re[b1]; re[b1] = tr;
          float ti = im[b]; im[b] = im[b1]; im[b1] = ti;
        }
      }
    }
  }
#pragma unroll
  for (int b = 0; b < 16; ++b) { Ure[f][b][col] = re[b]; Uim[f][b][col] = im[b]; }
  __syncthreads();

  // Pre-swizzled quantum B operands.
  // Logical B (32 x 16), per filter f, part p:
  //   p=0 (-> out_re): B[k][N] = k<16 ?  Ure[f][N][k] : -Uim[f][N][k-16]
  //   p=1 (-> out_im): B[k][N] = k<16 ?  Uim[f][N][k] :  Ure[f][N][k-16]
  float* Bq = ws + WS_BQ;
  for (int idx = t; idx < 2048; idx += 64) {   // idx = ((f*2+p)*8+c)*32 + lane
    const int ff = idx >> 9;
    const int rem = idx & 511;
    const int p = rem >> 8;
    const int c = (rem >> 5) & 7;
    const int lane = rem & 31;
    const int N = lane & 15;
    const int kx = 4 * c + ((lane >= 16) ? 2 : 0);
    auto val = [&](int k) -> float {
      if (k < 16) return (p == 0) ? Ure[ff][N][k] : Uim[ff][N][k];
      return (p == 0) ? -Uim[ff][N][k - 16] : Ure[ff][N][k - 16];
    };
    Bq[idx * 2 + 0] = val(kx);
    Bq[idx * 2 + 1] = val(kx + 1);
  }

  // Pre-swizzled W1 B operands: B[k][N] = W1[N][k]  (h = feat @ W1^T)
  float* Bw1 = ws + WS_BW1;
  for (int idx = t; idx < 2048; idx += 64) {   // idx = (ot*16+c)*32 + lane
    const int ot = idx >> 9;
    const int rem = idx & 511;
    const int c = rem >> 5;
    const int lane = rem & 31;
    const int N = ot * 16 + (lane & 15);
    const int kx = 4 * c + ((lane >= 16) ? 2 : 0);
    Bw1[idx * 2 + 0] = W1[N * 64 + kx];
    Bw1[idx * 2 + 1] = W1[N * 64 + kx + 1];
  }
}

// ---------------------------------------------------------------------------
// Kernel B: one wave per batch element. Each lane builds product-state psi for
// pixel (lane&15) fully in registers; WMMA f32 16x16x4 chains over K=32
// ([psi_re | psi_im]) per filter give out_re/out_im; <Z0> falls out of the C
// layout with a 16-lane xor reduction.
// ---------------------------------------------------------------------------
__global__ void __launch_bounds__(256)
quantum_feat(const float* __restrict__ x,      // (BATCH, 4, 16)
             const float* __restrict__ ws_bq,  // pre-swizzled Bq
             float* __restrict__ feat) {       // (BATCH, 64)
  const int lane = threadIdx.x & 31;
  const int wave = blockIdx.x * (blockDim.x >> 5) + (threadIdx.x >> 5);
  if (wave >= BATCHk) return;   // wave-uniform guard; EXEC all-ones below

  // Loop-invariant B operands -> registers (L2-resident, 16 KB total)
  v2f Bq[4][2][8];
#pragma unroll
  for (int f = 0; f < 4; ++f)
#pragma unroll
    for (int p = 0; p < 2; ++p)
#pragma unroll
      for (int c = 0; c < 8; ++c)
        Bq[f][p][c] = *(const v2f*)(ws_bq + ((((f * 2 + p) * 8 + c) * 32 + lane) * 2));

  const int batch = wave;
  const int pix = lane & 15;           // lanes l and l+16 build the same row
  const float* xb = x + batch * 64 + pix;
  float hc[4], hs[4];
#pragma unroll
  for (int i = 0; i < 4; ++i) {
    const float ang = xb[i * 16] * (PI_F * 0.5f);
    __sincosf(ang, &hs[i], &hc[i]);
  }

  // psi amplitudes: prod over bits, phase (-i)^popcount
  float cat[32];   // [0:16) = re, [16:32) = im
  const float rt[5] = {1.f, 0.f, -1.f, 0.f, 1.f};
  const float it[5] = {0.f, -1.f, 0.f, 1.f, 0.f};
#pragma unroll
  for (int b = 0; b < 16; ++b) {
    float p = ((b >> 3) & 1) ? hs[0] : hc[0];
    p *= ((b >> 2) & 1) ? hs[1] : hc[1];
    p *= ((b >> 1) & 1) ? hs[2] : hc[2];
    p *= (b & 1) ? hs[3] : hc[3];
    const int pc = __popc(b);          // compile-time per unrolled b
    cat[b]      = p * rt[pc];
    cat[16 + b] = p * it[pc];
  }

  // A operands, f32 16x4 layout: lanes<16 -> (K0,K1), lanes>=16 -> (K2,K3)
  const bool hi = lane >= 16;
  v2f A[8];
#pragma unroll
  for (int c = 0; c < 8; ++c) {
    A[c].x = hi ? cat[4 * c + 2] : cat[4 * c + 0];
    A[c].y = hi ? cat[4 * c + 3] : cat[4 * c + 1];
  }

  const float sgn = (lane & 8) ? -1.f : 1.f;   // s_N for N = lane&15
  float* fout = feat + batch * 64;
#pragma unroll
  for (int f = 0; f < 4; ++f) {
    v8f accRe = {0.f, 0.f, 0.f, 0.f, 0.f, 0.f, 0.f, 0.f};
    v8f accIm = {0.f, 0.f, 0.f, 0.f, 0.f, 0.f, 0.f, 0.f};
#pragma unroll
    for (int c = 0; c < 8; ++c) {
      accRe = __builtin_amdgcn_wmma_f32_16x16x4_f32(
          false, A[c], false, Bq[f][0][c], (short)0, accRe, false, false);
      accIm = __builtin_amdgcn_wmma_f32_16x16x4_f32(
          false, A[c], false, Bq[f][1][c], (short)0, accIm, false, false);
    }
    // q[M] = sum_N s_N * |out|^2 ; VGPR j holds M=j (lanes 0-15) / M=j+8
#pragma unroll
    for (int j = 0; j < 8; ++j) {
      float tv = sgn * (accRe[j] * accRe[j] + accIm[j] * accIm[j]);
      tv += __shfl_xor(tv, 1, 32);
      tv += __shfl_xor(tv, 2, 32);
      tv += __shfl_xor(tv, 4, 32);
      tv += __shfl_xor(tv, 8, 32);
      if (lane == 0)       fout[f * 16 + j]     = tv;  // pixel j
      else if (lane == 16) fout[f * 16 + 8 + j] = tv;  // pixel j+8
    }
  }
}

// ---------------------------------------------------------------------------
// Kernel C: h = relu(feat @ W1^T + b1), one wave per 16-batch tile, WMMA f32
// 16x16x4 chained over K=64, 4 output tiles of 16.
// ---------------------------------------------------------------------------
__global__ void __launch_bounds__(256)
mlp1(const float* __restrict__ feat, const float* __restrict__ ws_bw1,
     const float* __restrict__ b1, float* __restrict__ h) {
  const int lane = threadIdx.x & 31;
  const int wave = blockIdx.x * (blockDim.x >> 5) + (threadIdx.x >> 5);
  if (wave >= BATCHk / 16) return;

  v2f Bw[4][16];
#pragma unroll
  for (int ot = 0; ot < 4; ++ot)
#pragma unroll
    for (int c = 0; c < 16; ++c)
      Bw[ot][c] = *(const v2f*)(ws_bw1 + (((ot * 16 + c) * 32 + lane) * 2));

  const int tb = wave * 16;
  const int row = tb + (lane & 15);
  const int off = (lane >= 16) ? 2 : 0;
  v2f A[16];
#pragma unroll
  for (int c = 0; c < 16; ++c)
    A[c] = *(const v2f*)(feat + row * 64 + 4 * c + off);

#pragma unroll
  for (int ot = 0; ot < 4; ++ot) {
    v8f acc = {0.f, 0.f, 0.f, 0.f, 0.f, 0.f, 0.f, 0.f};
#pragma unroll
    for (int c = 0; c < 16; ++c)
      acc = __builtin_amdgcn_wmma_f32_16x16x4_f32(
          false, A[c], false, Bw[ot][c], (short)0, acc, false, false);
    const float bias = b1[ot * 16 + (lane & 15)];
#pragma unroll
    for (int j = 0; j < 8; ++j) {
      const int m = j + ((lane < 16) ? 0 : 8);
      float v = acc[j] + bias;
      v = v > 0.f ? v : 0.f;
      h[(tb + m) * 64 + ot * 16 + (lane & 15)] = v;
    }
  }
}

// ---------------------------------------------------------------------------
// Kernel D: batchnorm statistics (single block) folded with gamma/beta/W2/b2
// into 64 coefficients + 1 constant:  out[n] = sum_o h[n,o]*coef[o] + dconst.
// ---------------------------------------------------------------------------
__global__ void __launch_bounds__(1024)
bn_stats(const float* __restrict__ h, const float* __restrict__ gamma,
         const float* __restrict__ beta, const float* __restrict__ W2,
         const float* __restrict__ b2, float* __restrict__ coef,
         float* __restrict__ dconst) {
  __shared__ float ssum[16][64];
  __shared__ float ssq[16][64];
  __shared__ float sred[64];
  const int t = threadIdx.x;
  const int col = t & 63, seg = t >> 6;
  float s = 0.f, s2 = 0.f;
  const int rows = BATCHk / 16;
  for (int i = 0; i < rows; ++i) {
    const float v = h[(seg * rows + i) * 64 + col];
    s += v; s2 += v * v;
  }
  ssum[seg][col] = s; ssq[seg][col] = s2;
  __syncthreads();
  if (t < 64) {
    float S = 0.f, S2 = 0.f;
#pragma unroll
    for (int g = 0; g < 16; ++g) { S += ssum[g][t]; S2 += ssq[g][t]; }
    const float mu  = S / (float)BATCHk;
    const float var = S2 / (float)BATCHk - mu * mu;
    const float inv = rsqrtf(var + 1e-5f);
    const float gm = gamma[t], w2 = W2[t];
    coef[t] = gm * inv * w2;
    sred[t] = (beta[t] - mu * inv * gm) * w2;
  }
  __syncthreads();
  if (t == 0) {
    float d = b2[0];
#pragma unroll
    for (int o = 0; o < 64; ++o) d += sred[o];
    *dconst = d;
  }
}

// ---------------------------------------------------------------------------
// Kernel E: one wave per row, 64-dim dot + constant.
// ---------------------------------------------------------------------------
__global__ void __launch_bounds__(256)
final_out(const float* __restrict__ h, const float* __restrict__ coef,
          const float* __restrict__ dconst, float* __restrict__ out) {
  const int lane = threadIdx.x & 31;
  const int row = blockIdx.x * (blockDim.x >> 5) + (threadIdx.x >> 5);
  if (row >= BATCHk) return;
  float v = h[row * 64 + lane] * coef[lane] +
            h[row * 64 + 32 + lane] * coef[lane + 32];
  v += __shfl_xor(v, 1, 32);
  v += __shfl_xor(v, 2, 32);
  v += __shfl_xor(v, 4, 32);
  v += __shfl_xor(v, 8, 32);
  v += __shfl_xor(v, 16, 32);
  if (lane == 0) out[row] = v + *dconst;
}

// ---------------------------------------------------------------------------
extern "C" void kernel_launch(void* const* d_in, const int* in_sizes, int n_in,
                              void* d_out, int out_size, void* d_ws, size_t ws_size,
                              hipStream_t stream) {
  const float* x     = (const float*)d_in[0];
  const float* qw    = (const float*)d_in[1];
  const float* W1    = (const float*)d_in[2];
  const float* b1    = (const float*)d_in[3];
  const float* gamma = (const float*)d_in[4];
  const float* beta  = (const float*)d_in[5];
  const float* W2    = (const float*)d_in[6];
  const float* b2    = (const float*)d_in[7];
  float* ws   = (float*)d_ws;
  float* out  = (float*)d_out;
  float* Bq   = ws + WS_BQ;
  float* Bw1  = ws + WS_BW1;
  float* feat = ws + WS_FEAT;
  float* h    = ws + WS_H;
  float* coef = ws + WS_COEF;
  float* dcon = ws + WS_DCONST;

  hipLaunchKernelGGL(build_tables, dim3(1), dim3(64), 0, stream, qw, W1, ws);
  hipLaunchKernelGGL(quantum_feat, dim3((BATCHk * 32) / 256), dim3(256), 0,
                     stream, x, Bq, feat);
  hipLaunchKernelGGL(mlp1, dim3(((BATCHk / 16) * 32) / 256), dim3(256), 0,
                     stream, feat, Bw1, b1, h);
  hipLaunchKernelGGL(bn_stats, dim3(1), dim3(1024), 0, stream, h, gamma, beta,
                     W2, b2, coef, dcon);
  hipLaunchKernelGGL(final_out, dim3((BATCHk * 32) / 256), dim3(256), 0,
                     stream, h, coef, dcon, out);
}